// MaxUnpooling2D_75840532513303
// MI455X (gfx1250) — compile-verified
//
#include <hip/hip_runtime.h>
#include <cstdint>

// MaxUnpooling2D scatter-add, MI455X (gfx1250, wave32).
//   updates: (16,128,128,64) f32   mask: same shape i32, values in [0, OH*OW*C)
//   out:     (16,256,256,64) f32 zero-init, out[b][(mask & ~63) + (e & 63)] += updates[e]
// All sizes are powers of two:
//   H*W*C   = 2^20  (per-batch input volume)
//   OH*OW*C = 2^22  (per-batch output volume)
//   C       = 64
//
// Memory-bound scatter: ~0.5 GiB total traffic -> ~25 us floor at 23.3 TB/s.
// No matmul structure -> WMMA inapplicable; the CDNA5 lever used here is the
// async LDS->global store path (ASYNCcnt) for the zero pass, plus NT load
// hints so the 128 MiB of read-once inputs don't evict the atomic targets
// from the 192 MB L2 (output = 256 MiB, mostly L2-resident after zeroing).

#define IN_PER_BATCH_LOG2  20
#define OUT_PER_BATCH_LOG2 22
#define C_MASK 63

typedef __attribute__((ext_vector_type(4))) float v4f;
typedef __attribute__((ext_vector_type(4))) int   v4i;

// ---------------------------------------------------------------------------
// Pass 1: zero-fill d_out via the CDNA5 async LDS->global path.
// Each thread owns 16 bytes of zeros in LDS and streams them out with
// global_store_async_from_lds_b128 (ASYNCcnt-tracked, no VGPR data).
// Zeros are stored with default (RT) temporal hint on purpose: the atomic
// pass wants these lines resident in L2.
// ---------------------------------------------------------------------------
__global__ void unpool_zero_async_kernel(float* __restrict__ out, int n4 /* # of float4 */) {
    __shared__ v4f zbuf[256];
    const int t = threadIdx.x;
    zbuf[t] = (v4f){0.f, 0.f, 0.f, 0.f};
    __syncthreads();  // s_wait_dscnt + barrier: LDS zeros visible before async reads

    // Flat-pointer low 32 bits == workgroup-relative LDS byte offset
    // (LDS aperture keeps the offset in addr[31:0]).
    const uint32_t lds_off = (uint32_t)(uintptr_t)(&zbuf[t]);

    v4f* o4 = reinterpret_cast<v4f*>(out);
    long long idx    = (long long)blockIdx.x * blockDim.x + t;
    long long stride = (long long)gridDim.x * blockDim.x;

    for (; idx < n4; idx += stride) {
        uint64_t gaddr = (uint64_t)(uintptr_t)(o4 + idx);
        // copies 16 bytes from the LDS address in lds_off to the global address in gaddr, tracked by ASYNCcnt
        asm volatile("global_store_async_from_lds_b128 %0, %1, off"
                     :: "v"(gaddr), "v"(lds_off)
                     : "memory");
    }
    // Drain the async queue (s_endpgm also wait-idles, but be explicit).
    asm volatile("s_wait_asynccnt 0" ::: "memory");
}

// ---------------------------------------------------------------------------
// Pass 2: scatter-add. One thread = 4 consecutive channel elements:
//   2x global_load_b128 (TH=NT, read-once streams) in,
//   4x global_atomic_add_f32 (no-return form, STOREcnt-tracked) out.
// ---------------------------------------------------------------------------
__global__ void unpool_scatter_kernel(const v4f* __restrict__ up4,
                                      const v4i* __restrict__ mk4,
                                      float* __restrict__ out, int n4) {
    const int i = blockIdx.x * blockDim.x + threadIdx.x;
    if (i >= n4) return;

    // Non-temporal: inputs are streamed exactly once; keep L2 for the
    // atomic-target output lines instead.
    const v4f u = __builtin_nontemporal_load(up4 + i);
    const v4i m = __builtin_nontemporal_load(mk4 + i);

    const int e    = i << 2;                      // flat element index (< 2^24)
    const int b    = e >> IN_PER_BATCH_LOG2;      // batch
    const int c    = e & C_MASK;                  // channel of element 0 (mult. of 4)
    const int base = b << OUT_PER_BATCH_LOG2;     // batch offset in output (< 2^26)

    // Reference semantics: y/x come from mask, channel comes from position:
    //   dst = (mask / C) * C + c  ==  (mask & ~63) + c     (C == 64)
    // Return value discarded -> compiler emits global_atomic_add_f32 (no RTN).
    atomicAdd(&out[base + (m.x & ~C_MASK) + c + 0], u.x);
    atomicAdd(&out[base + (m.y & ~C_MASK) + c + 1], u.y);
    atomicAdd(&out[base + (m.z & ~C_MASK) + c + 2], u.z);
    atomicAdd(&out[base + (m.w & ~C_MASK) + c + 3], u.w);
}

extern "C" void kernel_launch(void* const* d_in, const int* in_sizes, int n_in,
                              void* d_out, int out_size, void* d_ws, size_t ws_size,
                              hipStream_t stream) {
    const float* updates = (const float*)d_in[0];   // f32, 16*128*128*64
    const int*   mask    = (const int*)d_in[1];     // i32, same count
    float*       out     = (float*)d_out;           // f32, 16*256*256*64

    // Pass 1: zero output (256 MiB) with async B128 stores, grid-stride.
    const int n_out4 = out_size >> 2;               // 2^24 float4 chunks
    unpool_zero_async_kernel<<<4096, 256, 0, stream>>>(out, n_out4);

    // Pass 2: scatter-add (same stream -> ordered after the zero pass).
    const int n4 = in_sizes[0] >> 2;                // 2^22 float4/int4 chunks
    unpool_scatter_kernel<<<(n4 + 255) / 256, 256, 0, stream>>>(
        (const v4f*)updates, (const v4i*)mask, out, n4);
}